// Inter_domain_loss_51754355917272
// MI455X (gfx1250) — compile-verified
//
#include <hip/hip_runtime.h>

typedef __attribute__((ext_vector_type(2))) float f32x2;
typedef __attribute__((ext_vector_type(8))) float f32x8;

#define NDIM 4096
#define EPSV 1e-5f
#define ITERS 100
#define SINK_BLOCKS 256   // one wave (32 threads) per block, one 16-col group per wave

// ---------------------------------------------------------------------------
// Column means of a [B, N] matrix (coalesced: thread per column).
// ---------------------------------------------------------------------------
__global__ void mean_cols(const float* __restrict__ src, float* __restrict__ out, int B) {
  int j = blockIdx.x * blockDim.x + threadIdx.x;
  if (j >= NDIM) return;
  float s = 0.f;
  for (int b = 0; b < B; ++b) s += src[(size_t)b * NDIM + j];
  out[j] = s * (1.0f / (float)B);
}

// u = 1, barrier counter = 0 (must be reset every call: harness replays the graph)
__global__ void init_vec(float* __restrict__ u, unsigned* __restrict__ bar) {
  int i = blockIdx.x * blockDim.x + threadIdx.x;
  if (i < NDIM) u[i] = 1.0f;
  if (i == 0) *bar = 0u;
}

// ---------------------------------------------------------------------------
// AT = transpose(graph) + EPS  (LDS-tiled, conflict-free)
// ---------------------------------------------------------------------------
__global__ void transpose_eps(const float* __restrict__ g, float* __restrict__ at) {
  __shared__ float t[32][33];
  int x  = blockIdx.x * 32 + threadIdx.x;
  int y0 = blockIdx.y * 32 + threadIdx.y;
#pragma unroll
  for (int k = 0; k < 32; k += 8)
    t[threadIdx.y + k][threadIdx.x] = g[(size_t)(y0 + k) * NDIM + x];
  __syncthreads();
  int x2 = blockIdx.y * 32 + threadIdx.x;
  int y2 = blockIdx.x * 32 + threadIdx.y;
#pragma unroll
  for (int k = 0; k < 32; k += 8)
    at[(size_t)(y2 + k) * NDIM + x2] = t[threadIdx.x][threadIdx.y + k] + EPSV;
}

// ---------------------------------------------------------------------------
// Grid-wide barrier for the persistent Sinkhorn kernel.
// Monotonic counter; target = nblocks * phase. Device-scope acquire/release.
// ---------------------------------------------------------------------------
__device__ inline void grid_bar(unsigned* bar, unsigned target) {
  __threadfence();  // release our u/v writes to L2
  if (threadIdx.x == 0) {
    __hip_atomic_fetch_add(bar, 1u, __ATOMIC_RELEASE, __HIP_MEMORY_SCOPE_AGENT);
    while (__hip_atomic_load(bar, __ATOMIC_ACQUIRE, __HIP_MEMORY_SCOPE_AGENT) < target)
      __builtin_amdgcn_s_sleep(1);
  }
  __syncthreads();
  __threadfence();  // invalidate near caches so we see other WGPs' u/v writes
}

// ---------------------------------------------------------------------------
// y[j0+n] = 1 / ( sum_i M[i][j0+n] * x[i] + epsSum ),  n = 0..15, via WMMA.
//
// V_WMMA_F32_16X16X4_F32:  D(16x16) = A(16x4) * B(4x16) + C
//   B-tile = M[i0..i0+3][j0..j0+15]   (row-major chunks -> coalesced 64B rows)
//     lane n   (0..15): V0 = B[0][n] = M[i0+0][j0+n], V1 = B[1][n] = M[i0+1][j0+n]
//     lane 16+n       : V0 = B[2][n],                 V1 = B[3][n]
//   A-tile = x[i0..i0+3] replicated over the 16 M-rows
//     lanes 0..15 : V0 = x[i0],   V1 = x[i0+1]
//     lanes 16..31: V0 = x[i0+2], V1 = x[i0+3]
//   => every column n of D accumulates the full dot product for column j0+n.
// ---------------------------------------------------------------------------
__device__ inline void wmatvec(const float* __restrict__ M, const float* __restrict__ x,
                               float* __restrict__ y, float epsSum, int j0, int lane) {
  const int n    = lane & 15;
  const int koff = (lane >> 4) << 1;  // 0 for lanes 0-15, 2 for lanes 16-31
  f32x8 acc = {};
  const float* bptr = M + (size_t)koff * NDIM + (j0 + n);
  const float* xptr = x + koff;
#pragma unroll 8
  for (int i0 = 0; i0 < NDIM; i0 += 4) {
    f32x2 a, b;
    b[0] = bptr[0];
    b[1] = bptr[NDIM];
    a[0] = xptr[0];
    a[1] = xptr[1];
    acc = __builtin_amdgcn_wmma_f32_16x16x4_f32(false, a, false, b, (short)0, acc,
                                                false, false);
    bptr += (size_t)4 * NDIM;
    xptr += 4;
  }
  // D layout: lanes 0..15 hold column N=lane; VGPR0 row M=0. All rows equal.
  if (lane < 16) y[j0 + lane] = 1.0f / (acc[0] + epsSum);
}

// ---------------------------------------------------------------------------
// Persistent Sinkhorn: g = diag(u) (graph+EPS) diag(v)
//   v = 1/((graph+EPS)^T u) = 1/(graph^T u + EPS*sum(u))
//   u = 1/((graph+EPS)  v ) = 1/(AT^T v)          (EPS baked into AT)
// 256 waves, wave w owns columns [16w, 16w+16). Both 64MB matrices stay in L2.
// ---------------------------------------------------------------------------
__global__ void __launch_bounds__(32)
sinkhorn_persistent(const float* __restrict__ graph, const float* __restrict__ at,
                    float* __restrict__ u, float* __restrict__ v,
                    unsigned* __restrict__ bar) {
  const int lane = threadIdx.x;
  const int j0   = blockIdx.x * 16;
  unsigned phase = 0;
  for (int it = 0; it < ITERS; ++it) {
    // sum(u) for the EPS correction on the graph side (wave-local, redundant, cheap)
    float s = 0.f;
    for (int i = lane; i < NDIM; i += 32) s += u[i];
#pragma unroll
    for (int o = 16; o > 0; o >>= 1) s += __shfl_xor(s, o, 32);

    wmatvec(graph, u, v, EPSV * s, j0, lane);          // column normalize -> v
    grid_bar(bar, ++phase * (unsigned)SINK_BLOCKS);
    wmatvec(at, v, u, 0.0f, j0, lane);                 // row normalize -> u
    grid_bar(bar, ++phase * (unsigned)SINK_BLOCKS);
  }
}

// ---------------------------------------------------------------------------
// loss partials: sum_ij |mt_i - ms_j| * u_i * (graph_ij + EPS) * v_j
// ---------------------------------------------------------------------------
__global__ void final_partial(const float* __restrict__ g, const float* __restrict__ u,
                              const float* __restrict__ v, const float* __restrict__ ms,
                              const float* __restrict__ mt, float* __restrict__ part) {
  const size_t total = (size_t)NDIM * NDIM;
  float s = 0.f;
  for (size_t e = (size_t)blockIdx.x * blockDim.x + threadIdx.x; e < total;
       e += (size_t)gridDim.x * blockDim.x) {
    int i = (int)(e >> 12);
    int j = (int)(e & (NDIM - 1));
    float d = fabsf(mt[i] - ms[j]);
    s += d * u[i] * (g[e] + EPSV) * v[j];
  }
  __shared__ float red[256];
  red[threadIdx.x] = s;
  __syncthreads();
  for (int off = 128; off > 0; off >>= 1) {
    if (threadIdx.x < off) red[threadIdx.x] += red[threadIdx.x + off];
    __syncthreads();
  }
  if (threadIdx.x == 0) part[blockIdx.x] = red[0];
}

__global__ void sum_parts(const float* __restrict__ part, int n, float* __restrict__ out) {
  __shared__ float red[256];
  float s = 0.f;
  for (int i = threadIdx.x; i < n; i += 256) s += part[i];  // fixed order: deterministic
  red[threadIdx.x] = s;
  __syncthreads();
  for (int off = 128; off > 0; off >>= 1) {
    if (threadIdx.x < off) red[threadIdx.x] += red[threadIdx.x + off];
    __syncthreads();
  }
  if (threadIdx.x == 0) out[0] = red[0];
}

// ---------------------------------------------------------------------------
extern "C" void kernel_launch(void* const* d_in, const int* in_sizes, int n_in,
                              void* d_out, int out_size, void* d_ws, size_t ws_size,
                              hipStream_t stream) {
  (void)n_in; (void)out_size; (void)ws_size;
  const float* src   = (const float*)d_in[0];
  const float* trg   = (const float*)d_in[1];
  const float* graph = (const float*)d_in[2];
  // d_in[3], d_in[4] (embedding weights) are unused by the reference.
  const int B = in_sizes[0] / NDIM;  // 2048

  float* ws   = (float*)d_ws;
  float* AT   = ws;                                   // 4096*4096 f32 (64 MB)
  float* u    = ws + (size_t)NDIM * NDIM;             // 4096
  float* v    = u + NDIM;                             // 4096
  float* ms   = v + NDIM;                             // 4096
  float* mt   = ms + NDIM;                            // 4096
  float* part = mt + NDIM;                            // 4096 (1024 used)
  unsigned* bar = (unsigned*)(part + NDIM);

  mean_cols<<<16, 256, 0, stream>>>(src, ms, B);
  mean_cols<<<16, 256, 0, stream>>>(trg, mt, B);
  init_vec<<<16, 256, 0, stream>>>(u, bar);
  transpose_eps<<<dim3(128, 128), dim3(32, 8), 0, stream>>>(graph, AT);
  sinkhorn_persistent<<<SINK_BLOCKS, 32, 0, stream>>>(graph, AT, u, v, bar);
  final_partial<<<1024, 256, 0, stream>>>(graph, u, v, ms, mt, part);
  sum_parts<<<1, 256, 0, stream>>>(part, 1024, (float*)d_out);
}